// FM_79070347919991
// MI455X (gfx1250) — compile-verified
//
#include <hip/hip_runtime.h>
#include <hip/hip_bf16.h>
#include <math.h>

// FM forward: out = sigmoid(first + 0.5*sum_k(lin_k^2 - sq_k))
// Inputs (setup_inputs order):
//   d_in[0] dense_inputs : float32 [16384 x 13]
//   d_in[1] sparse_inputs: int32   [16384 x 26]   (values < 100000)
//   d_in[2] w0           : float32 [1]
//   d_in[3] w            : float32 [2600013 x 1]
//   d_in[4] V            : float32 [32 x 2600013]  (row-major, stride F)
// Output: float32 [16384]

#define BATCH      16384
#define NDENSE     13
#define NSPARSE    26
#define FEATN      100000
#define KDIM       32
#define FSTRIDE    2600013ll          // 26*100000 + 13
#define ROWS_PER_BLOCK 128
#define THREADS    128
#define NWAVES     4
#define LDS_STRIDE 33                 // 33 is odd -> conflict-free row-major access

typedef __attribute__((ext_vector_type(2))) float v2f;
typedef __attribute__((ext_vector_type(8))) float v8f;

__global__ __launch_bounds__(THREADS)
void fm_fwd_kernel(const float* __restrict__ dense,
                   const int*   __restrict__ sparse,
                   const float* __restrict__ w0,
                   const float* __restrict__ w,
                   const float* __restrict__ V,
                   float*       __restrict__ out)
{
    __shared__ float lds_lin[ROWS_PER_BLOCK * LDS_STRIDE];
    __shared__ float lds_sq [ROWS_PER_BLOCK * LDS_STRIDE];

    const int tid      = threadIdx.x;
    const int lane     = tid & 31;
    // wave id is uniform per wave; make that visible to the compiler so the
    // tile loop is straight-line (EXEC stays all-ones for every WMMA).
    const int wave     = __builtin_amdgcn_readfirstlane(tid >> 5);
    const int half     = lane >> 4;   // 0: K pair {kk,kk+1}; 1: K pair {kk+2,kk+3}
    const int l16      = lane & 15;
    const int blockRow = blockIdx.x * ROWS_PER_BLOCK;

    // ---------------- Stage 1: dense GEMM tiles via V_WMMA_F32_16X16X4_F32 ----
    // D_lin = Xd(128x16pad) @ Vd^T(16pad x 32); D_sq = Xd.^2 @ (Vd.^2)^T
    // Tile map: wave w owns N-tile (w&1) and M-tiles {i*2 + (w>>1)}, i=0..3.
    const int nt = wave & 1;          // fixed N tile per wave -> B loads hoisted
    const int mh = wave >> 1;
    const int nCol = nt * 16 + l16;   // B: N index (latent k) = l16

    // B-matrix (and its elementwise square), loaded once per wave.
    // B[K=d][N=n] = V[n*F + d]; K layout: VGPR0=K{c0}, VGPR1=K{c0+1},
    // lanes 0-15 c0 = kk, lanes 16-31 c0 = kk+2. Zero-pad K=13..15 via select
    // after an always-in-bounds load (no exec-divergent predication).
    float bv [4][2];
    float bv2[4][2];
    #pragma unroll
    for (int s = 0; s < 4; ++s) {
        const int c0 = s * 4 + half * 2;
        const int c1 = c0 + 1;
        float b0 = V[(size_t)nCol * FSTRIDE + c0];   // c0 <= 15: in bounds
        float b1 = V[(size_t)nCol * FSTRIDE + c1];
        b0 = (c0 < NDENSE) ? b0 : 0.0f;
        b1 = (c1 < NDENSE) ? b1 : 0.0f;
        bv [s][0] = b0;      bv [s][1] = b1;
        bv2[s][0] = b0 * b0; bv2[s][1] = b1 * b1;
    }

    #pragma unroll
    for (int i = 0; i < 4; ++i) {
        const int mt   = i * 2 + mh;                 // M tile (16 rows)
        const int rowA = blockRow + mt * 16 + l16;   // A: both lane halves map M=l16
        const float* drow = dense + (size_t)rowA * NDENSE;

        v8f clin = {};
        v8f csq  = {};
        #pragma unroll
        for (int s = 0; s < 4; ++s) {
            const int c0 = s * 4 + half * 2;
            const int c1 = c0 + 1;
            // Clamp index (stay in bounds), then select zero: plain v_cndmask,
            // no EXEC manipulation around the WMMA.
            float a0 = drow[(c0 < NDENSE) ? c0 : 0];
            float a1 = drow[(c1 < NDENSE) ? c1 : 0];
            a0 = (c0 < NDENSE) ? a0 : 0.0f;
            a1 = (c1 < NDENSE) ? a1 : 0.0f;

            v2f A  = { a0, a1 };
            v2f B  = { bv[s][0], bv[s][1] };
            clin = __builtin_amdgcn_wmma_f32_16x16x4_f32(
                       false, A, false, B, (short)0, clin, false, false);

            v2f A2 = { a0 * a0, a1 * a1 };
            v2f B2 = { bv2[s][0], bv2[s][1] };
            csq  = __builtin_amdgcn_wmma_f32_16x16x4_f32(
                       false, A2, false, B2, (short)0, csq, false, false);
        }

        // C/D layout: VGPR v, lanes 0-15 -> (M=v, N=lane); lanes 16-31 -> (M=v+8)
        #pragma unroll
        for (int v = 0; v < 8; ++v) {
            const int rl = mt * 16 + v + half * 8;   // local row
            const int cc = nt * 16 + l16;            // latent index
            lds_lin[rl * LDS_STRIDE + cc] = clin[v];
            lds_sq [rl * LDS_STRIDE + cc] = csq[v];
        }
    }
    __syncthreads();

    // ---------------- Stage 2: per-thread sparse gather + reduction -----------
    const int r = blockRow + tid;

    float lin[KDIM], sq[KDIM];
    #pragma unroll
    for (int k = 0; k < KDIM; ++k) {
        lin[k] = lds_lin[tid * LDS_STRIDE + k];
        sq[k]  = lds_sq [tid * LDS_STRIDE + k];
    }

    // First-order term (w[0..12] are wave-uniform -> scalar loads)
    float first = w0[0];
    #pragma unroll
    for (int d = 0; d < NDENSE; ++d)
        first = fmaf(dense[(size_t)r * NDENSE + d], w[d], first);

    const int* sp = sparse + (size_t)r * NSPARSE;

    // 26 features x 32 latent scattered gathers. Lanes = consecutive batch rows,
    // so each (j,k) step's 32 addresses fall inside one 400KB window of V ->
    // near-full cacheline utilization; total HBM traffic ~= |V| = 333MB.
    for (int j = 0; j < NSPARSE; ++j) {
        const size_t col = (size_t)NDENSE + (size_t)j * FEATN + (size_t)sp[j];
        first += w[col];
        const float* vp = V + col;
        #pragma unroll
        for (int k = 0; k < KDIM; ++k) {
            const float v = vp[(size_t)k * FSTRIDE];
            lin[k] += v;
            sq[k]   = fmaf(v, v, sq[k]);
        }
    }

    float second = 0.0f;
    #pragma unroll
    for (int k = 0; k < KDIM; ++k)
        second += lin[k] * lin[k] - sq[k];

    const float x = first + 0.5f * second;
    out[r] = 1.0f / (1.0f + __expf(-x));
}

extern "C" void kernel_launch(void* const* d_in, const int* in_sizes, int n_in,
                              void* d_out, int out_size, void* d_ws, size_t ws_size,
                              hipStream_t stream) {
    (void)in_sizes; (void)n_in; (void)out_size; (void)d_ws; (void)ws_size;
    const float* dense  = (const float*)d_in[0];
    const int*   sparse = (const int*)  d_in[1];
    const float* w0     = (const float*)d_in[2];
    const float* w      = (const float*)d_in[3];
    const float* V      = (const float*)d_in[4];
    float*       out    = (float*)d_out;

    dim3 grid(BATCH / ROWS_PER_BLOCK);   // 128 blocks
    dim3 block(THREADS);                 // 128 threads = 4 wave32
    hipLaunchKernelGGL(fm_fwd_kernel, grid, block, 0, stream,
                       dense, sparse, w0, w, V, out);
}